// LightDreamerRSSM_661424963911
// MI455X (gfx1250) — compile-verified
//
#include <hip/hip_runtime.h>
#include <hip/hip_bf16.h>
#include <math.h>

#define Bsz  8
#define Vn   6
#define Ssz  32
#define Dd   128
#define Hh   128
#define OBSn 768
#define LN_EPSf 1e-5f
#define KL_EPSf 1e-5f

typedef __attribute__((ext_vector_type(2))) float v2f;
typedef __attribute__((ext_vector_type(4))) float v4f;
typedef __attribute__((ext_vector_type(8))) float v8f;

__device__ __forceinline__ float siluf(float x) { return x / (1.0f + expf(-x)); }
__device__ __forceinline__ float sigmf(float x) { return 1.0f / (1.0f + expf(-x)); }

// ---------------------------------------------------------------------------
// WMMA f32 16x16x4 core:  D = A(16x4) * B(4x16) + C   (fp32, RNE)
// ---------------------------------------------------------------------------
__device__ __forceinline__ v8f wmma4(v2f a, v2f b, v8f c) {
  return __builtin_amdgcn_wmma_f32_16x16x4_f32(false, a, false, b, (short)0, c,
                                               false, false);
}

// One 16x16 output tile of  A(8 real rows x K) @ W(K x ldw)  at columns [n0,n0+16).
// A is an LDS buffer with row stride lda (padded to avoid bank conflicts).
// Rows 8..15 of the A fragment are forced to zero branchlessly (EXEC stays full
// for WMMA as required by the ISA).
__device__ __forceinline__ v8f gemm_lds(const float* __restrict__ A, int lda,
                                        const float* __restrict__ W, int ldw,
                                        int K, int n0) {
  const int lane = threadIdx.x & 31;
  const int m    = lane & 15;
  const int hi   = lane >> 4;          // 0 or 1
  const float ms = (m < 8) ? 1.0f : 0.0f;
  const int  mr  = m & 7;
  const int  n   = n0 + (lane & 15);
  v8f c = {};
  for (int k = 0; k < K; k += 4) {
    const int ka = k + 2 * hi;
    v2f a, b;
    a.x = A[mr * lda + ka]     * ms;
    a.y = A[mr * lda + ka + 1] * ms;
    b.x = W[(size_t)ka       * ldw + n];
    b.y = W[(size_t)(ka + 1) * ldw + n];
    c = wmma4(a, b, c);
  }
  return c;
}

// Same but A lives in global memory (used for the pooled obs_t rows).
__device__ __forceinline__ v8f gemm_glb(const float* __restrict__ A, int lda,
                                        const float* __restrict__ W, int ldw,
                                        int K, int n0) {
  const int lane = threadIdx.x & 31;
  const int m    = lane & 15;
  const int hi   = lane >> 4;
  const float ms = (m < 8) ? 1.0f : 0.0f;
  const int  mr  = m & 7;
  const int  n   = n0 + (lane & 15);
  v8f c = {};
  for (int k = 0; k < K; k += 4) {
    const int ka = k + 2 * hi;
    v2f a, b;
    a.x = A[(size_t)mr * lda + ka]     * ms;
    a.y = A[(size_t)mr * lda + ka + 1] * ms;
    b.x = W[(size_t)ka       * ldw + n];
    b.y = W[(size_t)(ka + 1) * ldw + n];
    c = wmma4(a, b, c);
  }
  return c;
}

// Store the 8 real rows (batch) of a 16x16 D tile to LDS. Lanes 0..15 hold
// rows 0..7 at column n0+lane; lanes 16..31 hold the zero pad rows -> skipped.
__device__ __forceinline__ void store_tile(float* __restrict__ buf, int ld,
                                           int n0, v8f d) {
  const int lane = threadIdx.x & 31;
  if (lane < 16) {
    const int n = n0 + lane;
#pragma unroll
    for (int i = 0; i < 8; ++i) buf[i * ld + n] = d[i];
  }
}

// ---------------------------------------------------------------------------
// Kernel 1: spatial mean pooling (the only bandwidth-heavy part: ~415 MB).
// One block per output channel; nontemporal float4 streaming loads.
// Writes obs/tgt already transposed to (V, B, 768).
// ---------------------------------------------------------------------------
__global__ __launch_bounds__(128)
void pool_kernel(const float* __restrict__ sf0, const float* __restrict__ sf1,
                 const float* __restrict__ tf0, const float* __restrict__ tf1,
                 float* __restrict__ obs, float* __restrict__ tgt) {
  __shared__ float red[128];
  const int tid  = threadIdx.x;
  int cid        = blockIdx.x;
  const int half = Bsz * Vn * (256 + 512);   // 36864 channels per (student|teacher)

  float* dst = obs;
  const float* f0 = sf0;
  const float* f1 = sf1;
  if (cid >= half) { cid -= half; dst = tgt; f0 = tf0; f1 = tf1; }

  const float* src;
  int n4; float scale; size_t dstIdx;
  if (cid < Bsz * Vn * 256) {                        // f0: (B,V,256,32,88)
    const int c = cid & 255;
    const int v = (cid >> 8) % Vn;
    const int b = cid / (256 * Vn);
    src    = f0 + (size_t)((b * Vn + v) * 256 + c) * 2816;
    n4     = 2816 / 4;
    scale  = 1.0f / 2816.0f;
    dstIdx = (size_t)(v * Bsz + b) * OBSn + c;
  } else {                                           // f1: (B,V,512,16,44)
    const int r = cid - Bsz * Vn * 256;
    const int c = r & 511;
    const int v = (r >> 9) % Vn;
    const int b = r / (512 * Vn);
    src    = f1 + (size_t)((b * Vn + v) * 512 + c) * 704;
    n4     = 704 / 4;
    scale  = 1.0f / 704.0f;
    dstIdx = (size_t)(v * Bsz + b) * OBSn + 256 + c;
  }

  const v4f* p = (const v4f*)src;
  float s = 0.0f;
  for (int i = tid; i < n4; i += 128) {
    v4f x = __builtin_nontemporal_load(p + i);       // stream, don't pollute L2
    s += x.x + x.y + x.z + x.w;
  }
  red[tid] = s;
  __syncthreads();
  for (int o = 64; o > 0; o >>= 1) {
    if (tid < o) red[tid] += red[tid + o];
    __syncthreads();
  }
  if (tid == 0) dst[dstIdx] = red[0] * scale;
}

// ---------------------------------------------------------------------------
// Kernel 2: the full 6-step RSSM scan in one persistent workgroup (8 waves).
// All activations + deter state live in LDS; every GEMM uses fp32 WMMA.
// ---------------------------------------------------------------------------
__global__ __launch_bounds__(256)
void rssm_kernel(const float* __restrict__ obs, const float* __restrict__ tgt,
                 const unsigned char* __restrict__ cam_mask,
                 const float* __restrict__ eps_prior, const float* __restrict__ eps_post,
                 const float* __restrict__ enc_w1, const float* __restrict__ enc_b1,
                 const float* __restrict__ ln_g,   const float* __restrict__ ln_b,
                 const float* __restrict__ enc_w2, const float* __restrict__ enc_b2,
                 const float* __restrict__ prior_w1, const float* __restrict__ prior_b1,
                 const float* __restrict__ prior_w2, const float* __restrict__ prior_b2,
                 const float* __restrict__ post_w1,  const float* __restrict__ post_b1,
                 const float* __restrict__ post_w2,  const float* __restrict__ post_b2,
                 const float* __restrict__ dec_w1,   const float* __restrict__ dec_b1,
                 const float* __restrict__ dec_w2,   const float* __restrict__ dec_b2,
                 const float* __restrict__ gru_wih,  const float* __restrict__ gru_whh,
                 const float* __restrict__ gru_bih,  const float* __restrict__ gru_bhh,
                 float* __restrict__ out) {
  // A-operand buffers use padded strides (stride%32==4) -> conflict-free
  // 16-lane A-fragment reads. ~60 KB total static LDS.
  __shared__ float deterB[8 * 132];
  __shared__ float tbuf [8 * 128];
  __shared__ float tbuf2[8 * 128];
  __shared__ float aB1  [8 * 132];
  __shared__ float aB2  [8 * 132];
  __shared__ float postIn[8 * 260];
  __shared__ float prB[8 * 64];
  __shared__ float poB[8 * 64];
  __shared__ float zB [8 * 36];
  __shared__ float decIn[8 * 164];
  __shared__ float giB[8 * 384];
  __shared__ float ghB[8 * 384];
  __shared__ float reconRow[8];
  __shared__ float klRow[8];

  const int tid  = threadIdx.x;
  const int wave = tid >> 5;
  const int lane = tid & 31;

  for (int i = tid; i < 8 * 132; i += 256) deterB[i] = 0.0f;   // deter = 0

  float rl = 0.0f, rs = 0.0f, klacc = 0.0f, kn = 0.0f;

  for (int t = 0; t < Vn; ++t) {
    if (tid < 8) { reconRow[tid] = 0.0f; klRow[tid] = 0.0f; }
    __syncthreads();

    // enc1: obs_t(8x768) @ enc_w1(768x128) -> tbuf     (one N-tile per wave)
    { v8f c = gemm_glb(obs + (size_t)t * 8 * OBSn, OBSn, enc_w1, Hh, OBSn, wave * 16);
      store_tile(tbuf, 128, wave * 16, c); }
    __syncthreads();

    // LayerNorm(+bias) + silu -> aB1   (wave w handles batch row w)
    {
      const int b = wave, c0 = lane * 4;
      float x[4]; float s = 0.0f, ss = 0.0f;
#pragma unroll
      for (int j = 0; j < 4; ++j) {
        x[j] = tbuf[b * 128 + c0 + j] + enc_b1[c0 + j];
        s += x[j]; ss += x[j] * x[j];
      }
      for (int o = 16; o > 0; o >>= 1) {
        s  += __shfl_xor(s,  o, 32);
        ss += __shfl_xor(ss, o, 32);
      }
      const float mu  = s * (1.0f / 128.0f);
      const float var = ss * (1.0f / 128.0f) - mu * mu;
      const float ri  = rsqrtf(var + LN_EPSf);
#pragma unroll
      for (int j = 0; j < 4; ++j) {
        float xn = (x[j] - mu) * ri * ln_g[c0 + j] + ln_b[c0 + j];
        aB1[b * 132 + c0 + j] = siluf(xn);
      }
    }
    __syncthreads();

    // enc2: aB1 @ enc_w2 -> tbuf
    { v8f c = gemm_lds(aB1, 132, enc_w2, Hh, Hh, wave * 16);
      store_tile(tbuf, 128, wave * 16, c); }
    __syncthreads();

    // obs_embed = silu(.+b2) -> postIn[:,128:256]; copy old deter -> postIn[:,0:128]
    for (int idx = tid; idx < 1024; idx += 256) {
      const int b = idx >> 7, c = idx & 127;
      postIn[b * 260 + 128 + c] = siluf(tbuf[b * 128 + c] + enc_b2[c]);
      postIn[b * 260 + c]       = deterB[b * 132 + c];
    }
    __syncthreads();

    // prior1: deter @ prior_w1 -> tbuf ; post1: postIn @ post_w1 -> tbuf2
    { v8f c = gemm_lds(deterB, 132, prior_w1, Hh, Dd, wave * 16);
      store_tile(tbuf, 128, wave * 16, c); }
    { v8f c = gemm_lds(postIn, 260, post_w1, Hh, Dd + Hh, wave * 16);
      store_tile(tbuf2, 128, wave * 16, c); }
    __syncthreads();

    for (int idx = tid; idx < 1024; idx += 256) {
      const int b = idx >> 7, c = idx & 127;
      aB1[b * 132 + c] = siluf(tbuf [b * 128 + c] + prior_b1[c]);
      aB2[b * 132 + c] = siluf(tbuf2[b * 128 + c] + post_b1[c]);
    }
    __syncthreads();

    // prior2 / post2 (N=64 each -> 4 tiles each; waves 0-3 / 4-7)
    if (wave < 4) { v8f c = gemm_lds(aB1, 132, prior_w2, 2 * Ssz, Hh, wave * 16);
                    store_tile(prB, 64, wave * 16, c); }
    else          { v8f c = gemm_lds(aB2, 132, post_w2, 2 * Ssz, Hh, (wave - 4) * 16);
                    store_tile(poB, 64, (wave - 4) * 16, c); }
    __syncthreads();

    // reparameterize + mask-select z + KL row sums  (one (b,s) per thread)
    {
      const int b = tid >> 5, s = tid & 31;
      const float pm  = prB[b * 64 + s]      + prior_b2[s];
      const float plv = prB[b * 64 + 32 + s] + prior_b2[32 + s];
      const float qm  = poB[b * 64 + s]      + post_b2[s];
      const float qlv = poB[b * 64 + 32 + s] + post_b2[32 + s];
      const float epr = eps_prior[((size_t)t * Bsz + b) * Ssz + s];
      const float epo = eps_post [((size_t)t * Bsz + b) * Ssz + s];
      const float zpr = pm + epr * expf(0.5f * plv);
      const float zpo = qm + epo * expf(0.5f * qlv);
      const float mb  = cam_mask[b * Vn + t] ? 1.0f : 0.0f;
      zB[b * 36 + s] = (mb > 0.0f) ? zpr : zpo;
      const float vq = fmaxf(expf(qlv), KL_EPSf);
      const float vp = fmaxf(expf(plv), KL_EPSf);
      const float dmu = qm - pm;
      atomicAdd(&klRow[b], 0.5f * ((vq + dmu * dmu) / vp - 1.0f + plv - qlv));
    }
    __syncthreads();

    // GRU gemms: gi = z(8x32)@gru_wih(32x384); gh = deter(8x128)@gru_whh(128x384)
    for (int j = 0; j < 6; ++j) {
      const int tsk = wave + 8 * j;                      // 48 tile tasks, balanced
      if (tsk < 24) {
        v8f c = gemm_lds(zB, 36, gru_wih, 3 * Dd, Ssz, tsk * 16);
        store_tile(giB, 384, tsk * 16, c);
      } else {
        v8f c = gemm_lds(deterB, 132, gru_whh, 3 * Dd, Dd, (tsk - 24) * 16);
        store_tile(ghB, 384, (tsk - 24) * 16, c);
      }
    }
    __syncthreads();

    // GRU gates + deter update; stage dec_in = [deter_new | z]
    for (int idx = tid; idx < 1024; idx += 256) {
      const int b = idx >> 7, c = idx & 127;
      const float ir  = giB[b * 384 + c]       + gru_bih[c];
      const float iz  = giB[b * 384 + 128 + c] + gru_bih[128 + c];
      const float inn = giB[b * 384 + 256 + c] + gru_bih[256 + c];
      const float hr  = ghB[b * 384 + c]       + gru_bhh[c];
      const float hz  = ghB[b * 384 + 128 + c] + gru_bhh[128 + c];
      const float hn  = ghB[b * 384 + 256 + c] + gru_bhh[256 + c];
      const float r  = sigmf(ir + hr);
      const float u  = sigmf(iz + hz);
      const float nn = tanhf(inn + r * hn);
      const float dn = (1.0f - u) * nn + u * deterB[b * 132 + c];
      deterB[b * 132 + c] = dn;
      decIn[b * 164 + c]  = dn;
    }
    { const int b = tid >> 5, s = tid & 31;
      decIn[b * 164 + 128 + s] = zB[b * 36 + s]; }
    __syncthreads();

    // dec1: decIn(8x160) @ dec_w1 -> silu -> aB1
    { v8f c = gemm_lds(decIn, 164, dec_w1, Hh, Dd + Ssz, wave * 16);
      store_tile(tbuf, 128, wave * 16, c); }
    __syncthreads();
    for (int idx = tid; idx < 1024; idx += 256) {
      const int b = idx >> 7, c = idx & 127;
      aB1[b * 132 + c] = siluf(tbuf[b * 128 + c] + dec_b1[c]);
    }
    __syncthreads();

    // dec2: aB1(8x128) @ dec_w2(128x768) fused with squared-error reduction
    {
      float acc[8];
#pragma unroll
      for (int i = 0; i < 8; ++i) acc[i] = 0.0f;
      for (int j = 0; j < 6; ++j) {
        const int n0 = (wave + 8 * j) * 16;
        v8f c = gemm_lds(aB1, 132, dec_w2, OBSn, Hh, n0);
        if (lane < 16) {
          const int n = n0 + lane;
          const float bn = dec_b2[n];
#pragma unroll
          for (int i = 0; i < 8; ++i) {
            const float d = c[i] + bn - tgt[((size_t)t * Bsz + i) * OBSn + n];
            acc[i] += d * d;
          }
        }
      }
      if (lane < 16) {
#pragma unroll
        for (int i = 0; i < 8; ++i) atomicAdd(&reconRow[i], acc[i]);
      }
    }
    __syncthreads();

    // per-step masked loss accumulation (scalar; thread 0 keeps running sums)
    if (tid == 0) {
      float msum = 0.0f, rsum = 0.0f, ksum = 0.0f;
      for (int b = 0; b < Bsz; ++b) {
        const float mb = cam_mask[b * Vn + t] ? 1.0f : 0.0f;
        msum += mb;
        rsum += (reconRow[b] * (1.0f / (float)OBSn)) * mb;
        ksum += klRow[b] * (1.0f - mb);
      }
      if (msum > 0.0f) { rl += rsum / fmaxf(msum, 1.0f); rs += 1.0f; }
      const float osum = (float)Bsz - msum;
      if (osum > 0.0f) { klacc += ksum / fmaxf(osum, 1.0f); kn += 1.0f; }
    }
    __syncthreads();
  }

  if (tid == 0) {
    out[0] = rl / fmaxf(rs, 1.0f);                    // rec_loss * 1.0
    out[1] = (klacc / fmaxf(kn, 1.0f)) * 1e-4f;       // kl_loss * 0.0001
  }
}

// ---------------------------------------------------------------------------
extern "C" void kernel_launch(void* const* d_in, const int* in_sizes, int n_in,
                              void* d_out, int out_size, void* d_ws, size_t ws_size,
                              hipStream_t stream) {
  (void)in_sizes; (void)n_in; (void)out_size; (void)ws_size;

  const float* sf0 = (const float*)d_in[0];
  const float* sf1 = (const float*)d_in[1];
  const float* tf0 = (const float*)d_in[2];
  const float* tf1 = (const float*)d_in[3];
  const unsigned char* mask = (const unsigned char*)d_in[4];  // jnp bool = 1 byte
  const float* eps_pr = (const float*)d_in[5];
  const float* eps_po = (const float*)d_in[6];

  float* obs  = (float*)d_ws;                               // (V,B,768)
  float* tgtb = obs + (size_t)Vn * Bsz * OBSn;              // (V,B,768)

  const int nblocks = 2 * Bsz * Vn * (256 + 512);           // 73728 channels
  pool_kernel<<<nblocks, 128, 0, stream>>>(sf0, sf1, tf0, tf1, obs, tgtb);

  rssm_kernel<<<1, 256, 0, stream>>>(
      obs, tgtb, mask, eps_pr, eps_po,
      (const float*)d_in[7],  (const float*)d_in[8],   // enc_w1, enc_b1
      (const float*)d_in[9],  (const float*)d_in[10],  // ln_g, ln_b
      (const float*)d_in[11], (const float*)d_in[12],  // enc_w2, enc_b2
      (const float*)d_in[13], (const float*)d_in[14],  // prior_w1, prior_b1
      (const float*)d_in[15], (const float*)d_in[16],  // prior_w2, prior_b2
      (const float*)d_in[17], (const float*)d_in[18],  // post_w1, post_b1
      (const float*)d_in[19], (const float*)d_in[20],  // post_w2, post_b2
      (const float*)d_in[21], (const float*)d_in[22],  // dec_w1, dec_b1
      (const float*)d_in[23], (const float*)d_in[24],  // dec_w2, dec_b2
      (const float*)d_in[25], (const float*)d_in[26],  // gru_wih, gru_whh
      (const float*)d_in[27], (const float*)d_in[28],  // gru_bih, gru_bhh
      (float*)d_out);
}